// CMC_38276748542205
// MI455X (gfx1250) — compile-verified
//
#include <hip/hip_runtime.h>
#include <stdint.h>

// CMC: per-token SAD vs cached I-frame token, conditional replacement.
// Memory-bound (~0.25 FLOP/byte) -> optimize data movement, not math.

typedef float v4f __attribute__((ext_vector_type(4)));
typedef int   vint4 __attribute__((vector_size(16)));   // matches builtin param type
typedef __attribute__((address_space(3))) vint4 lds_vint4;

#define SEQ_LEN   12608
#define HIDDEN    4096
#define V4_PER_ROW (HIDDEN / 4)          // 1024
#define IMG_START 35
#define PATCH_NUM 196
#define NUM_FRAMES 64
#define IMG_LEN   (NUM_FRAMES * PATCH_NUM)   // 12544
#define THRESH    (1.12f * 4096.0f)          // 4587.52f

#if __has_builtin(__builtin_amdgcn_global_load_async_to_lds_b128)
#define CMC_ASYNC 1
#else
#define CMC_ASYNC 0
#endif

#if CMC_ASYNC
__device__ __forceinline__ void cmc_wait_async0() {
#if __has_builtin(__builtin_amdgcn_s_wait_asynccnt)
  __builtin_amdgcn_s_wait_asynccnt(0);
#else
  asm volatile("s_wait_asynccnt 0" ::: "memory");
#endif
}
#endif

__global__ __launch_bounds__(256) void cmc_kernel(const float* __restrict__ x,
                                                  float* __restrict__ out) {
  const int row = blockIdx.x;
  const int tid = threadIdx.x;

  const v4f* __restrict__ src = (const v4f*)(x + (size_t)row * HIDDEN);
  v4f* __restrict__ dst       = (v4f*)(out + (size_t)row * HIDDEN);

  const int img = row - IMG_START;
  if (img < 0 || img >= IMG_LEN) {
    // Passthrough rows (64 of them): streamed copy, non-temporal both ways.
#pragma unroll
    for (int k = 0; k < 4; ++k) {
      const int idx = tid + k * 256;
      v4f v = __builtin_nontemporal_load(&src[idx]);
      __builtin_nontemporal_store(v, &dst[idx]);
    }
    return;
  }

  const int frame  = img / PATCH_NUM;
  const int patch  = img - frame * PATCH_NUM;
  const int iframe = (frame & ~3) + 3;                   // I-frame of this interval
  const int irow   = IMG_START + iframe * PATCH_NUM + patch;
  const float* __restrict__ ibase = x + (size_t)irow * HIDDEN;

  __shared__ float lds_i[HIDDEN];   // 16 KB staging for the I-frame token
  __shared__ float red[8];          // cross-wave reduction slots

  v4f p[4], iv[4];

#if CMC_ASYNC
  // gfx1250 async global->LDS staging of the I-frame row (RT hint: reused 4x).
#pragma unroll
  for (int k = 0; k < 4; ++k) {
    const int idx = tid + k * 256;                       // float4 index
    __builtin_amdgcn_global_load_async_to_lds_b128(
        (vint4*)(ibase + (size_t)idx * 4),
        (lds_vint4*)(uintptr_t)(lds_i + (size_t)idx * 4),
        0, 0);
  }
#endif

  // P token: read exactly once -> non-temporal, keep L2 for I-frame rows.
#pragma unroll
  for (int k = 0; k < 4; ++k) {
    const int idx = tid + k * 256;
    p[k] = __builtin_nontemporal_load(&src[idx]);
  }

#if CMC_ASYNC
  cmc_wait_async0();   // each thread reads back only its own LDS chunks
  const v4f* __restrict__ lv = (const v4f*)lds_i;
#pragma unroll
  for (int k = 0; k < 4; ++k) iv[k] = lv[tid + k * 256];
#else
  const v4f* __restrict__ isrc = (const v4f*)ibase;
#pragma unroll
  for (int k = 0; k < 4; ++k) iv[k] = isrc[tid + k * 256];
#endif

  // Per-thread partial SAD over this thread's 16 elements.
  float s = 0.f;
#pragma unroll
  for (int k = 0; k < 4; ++k) {
    s += fabsf(p[k].x - iv[k].x);
    s += fabsf(p[k].y - iv[k].y);
    s += fabsf(p[k].z - iv[k].z);
    s += fabsf(p[k].w - iv[k].w);
  }

  // wave32 butterfly reduction.
#pragma unroll
  for (int off = 16; off > 0; off >>= 1)
    s += __shfl_xor(s, off, 32);

  const int lane = tid & 31;
  const int wave = tid >> 5;
  if (lane == 0) red[wave] = s;
  __syncthreads();

  float sad = 0.f;
#pragma unroll
  for (int w = 0; w < 8; ++w) sad += red[w];

  const bool replace = sad < THRESH;

  // Output row streamed once -> non-temporal stores.
#pragma unroll
  for (int k = 0; k < 4; ++k) {
    const int idx = tid + k * 256;
    __builtin_nontemporal_store(replace ? iv[k] : p[k], &dst[idx]);
  }
}

extern "C" void kernel_launch(void* const* d_in, const int* in_sizes, int n_in,
                              void* d_out, int out_size, void* d_ws, size_t ws_size,
                              hipStream_t stream) {
  (void)in_sizes; (void)n_in; (void)out_size; (void)d_ws; (void)ws_size;
  const float* x = (const float*)d_in[0];
  float* out = (float*)d_out;
  cmc_kernel<<<dim3(SEQ_LEN), dim3(256), 0, stream>>>(x, out);
}